// RotatedBboxesLoss_89816356094119
// MI455X (gfx1250) — compile-verified
//
#include <hip/hip_runtime.h>
#include <hip/hip_bf16.h>
#include <math.h>

typedef __attribute__((ext_vector_type(2))) float v2f;
typedef __attribute__((ext_vector_type(8))) float v8f;

#define NUM_CLASSES 80
#define REG_MAX     16
#define L_DIM       21504
#define DEG2RAD     0.017453292519943295f

// ---------------- Kernel 0: zero the two loss accumulators ----------------
__global__ void k_zero(float* acc) {
    if (threadIdx.x < 2) acc[threadIdx.x] = 0.0f;
}

// ---------------- Kernel 1: bbox_weight = target_scores.sum(-1) -----------
// One wave per 16-row tile. A = 16x4 f32 tile of scores, B = ones(4x16),
// accumulate D += A*B over 20 chunks -> every column of D holds the row sum.
// A layout (ISA 7.12.2, 32-bit A 16x4): lanes 0-15 hold K=0,1 (VGPR0,1) for
// M=lane; lanes 16-31 hold K=2,3 for M=lane-16.
// D layout: lane 0 VGPR j -> (M=j, N=0); lane 16 VGPR j -> (M=8+j, N=0).
__global__ void k_rowsum_wmma(const float* __restrict__ scores,
                              float* __restrict__ out, int ntiles) {
    int wave = (int)((blockIdx.x * blockDim.x + threadIdx.x) >> 5);
    int lane = (int)(threadIdx.x & 31u);
    if (wave >= ntiles) return;                 // wave-uniform: EXEC stays all-ones
    int m     = lane & 15;
    int khalf = (lane >> 4) << 1;               // 0 for lanes 0-15, 2 for 16-31
    const float* rowp = scores + (size_t)(wave * 16 + m) * NUM_CLASSES;

    v2f b; b.x = 1.0f; b.y = 1.0f;              // ones matrix (layout-independent)
    v8f c = {};
#pragma unroll
    for (int kc = 0; kc < NUM_CLASSES / 4; ++kc) {
        v2f a;
        a.x = rowp[kc * 4 + khalf + 0];
        a.y = rowp[kc * 4 + khalf + 1];
        c = __builtin_amdgcn_wmma_f32_16x16x4_f32(false, a, false, b,
                                                  (short)0, c, false, false);
    }
    int base = wave * 16;
    if (lane == 0) {
        out[base +  0] = c[0]; out[base +  1] = c[1];
        out[base +  2] = c[2]; out[base +  3] = c[3];
        out[base +  4] = c[4]; out[base +  5] = c[5];
        out[base +  6] = c[6]; out[base +  7] = c[7];
    } else if (lane == 16) {
        out[base +  8] = c[0]; out[base +  9] = c[1];
        out[base + 10] = c[2]; out[base + 11] = c[3];
        out[base + 12] = c[4]; out[base + 13] = c[5];
        out[base + 14] = c[6]; out[base + 15] = c[7];
    }
}

// ---------------- Device helpers for rotated IoU ---------------------------
__device__ __forceinline__ void box2corners(float x, float y, float w, float h,
                                            float a, float* cx, float* cy) {
    const float txs[4] = { 0.5f, -0.5f, -0.5f,  0.5f };
    const float tys[4] = { 0.5f,  0.5f, -0.5f, -0.5f };
    float c = __cosf(a), s = __sinf(a);
#pragma unroll
    for (int k = 0; k < 4; ++k) {
        float vx = txs[k] * w, vy = tys[k] * h;
        cx[k] = vx * c - vy * s + x;
        cy[k] = vx * s + vy * c + y;
    }
}

__device__ __forceinline__ bool corner_in_box(float qx, float qy,
                                              const float* cx, const float* cy) {
    float ax = cx[0], ay = cy[0];
    float abx = cx[1] - ax, aby = cy[1] - ay;
    float adx = cx[3] - ax, ady = cy[3] - ay;
    float apx = qx - ax,   apy = qy - ay;
    float pab = (apx * abx + apy * aby) / fmaxf(abx * abx + aby * aby, 1e-12f);
    float pad = (apx * adx + apy * ady) / fmaxf(adx * adx + ady * ady, 1e-12f);
    const float e = 1e-6f;
    return (pab > -e) && (pab < 1.0f + e) && (pad > -e) && (pad < 1.0f + e);
}

// ---------------- Kernel 2: fused IoU + DFL loss ---------------------------
__global__ void k_main(const float* __restrict__ pred_dist,
                       const float* __restrict__ pred_bboxes,
                       const float* __restrict__ pred_angles,
                       const float* __restrict__ anchor_points,
                       const float* __restrict__ target_bboxes,
                       const float* __restrict__ target_angles,
                       const unsigned char* __restrict__ fg_mask,
                       const float* __restrict__ bbox_weight,
                       float* __restrict__ acc, int n) {
    int i = (int)(blockIdx.x * blockDim.x + threadIdx.x);
    float s_iou = 0.0f, s_dfl = 0.0f;

    if (i < n && fg_mask[i]) {
        const float* pd = pred_dist + (size_t)i * (4 * (REG_MAX + 1));
        __builtin_prefetch(pd, 0, 0);          // global_prefetch_b8: pull DFL row early
        float w = bbox_weight[i];

        // ---- rotated IoU ----
        float px = pred_bboxes[(size_t)i * 4 + 0], py = pred_bboxes[(size_t)i * 4 + 1];
        float pw = pred_bboxes[(size_t)i * 4 + 2], ph = pred_bboxes[(size_t)i * 4 + 3];
        float tx = target_bboxes[(size_t)i * 4 + 0], ty = target_bboxes[(size_t)i * 4 + 1];
        float tw = target_bboxes[(size_t)i * 4 + 2], th = target_bboxes[(size_t)i * 4 + 3];
        float pa = pred_angles[i]   * DEG2RAD;
        float ta = target_angles[i] * DEG2RAD;

        float c1x[4], c1y[4], c2x[4], c2y[4];
        box2corners(px, py, pw, ph, pa, c1x, c1y);
        box2corners(tx, ty, tw, th, ta, c2x, c2y);

        float vx[24], vy[24];
        bool  mk[24];
#pragma unroll
        for (int k = 0; k < 4; ++k) {
            vx[k]     = c1x[k]; vy[k]     = c1y[k];
            vx[4 + k] = c2x[k]; vy[4 + k] = c2y[k];
            mk[k]     = corner_in_box(c1x[k], c1y[k], c2x, c2y);
            mk[4 + k] = corner_in_box(c2x[k], c2y[k], c1x, c1y);
        }
#pragma unroll
        for (int ie = 0; ie < 4; ++ie) {
            float p0x = c1x[ie], p0y = c1y[ie];
            float d1x = c1x[(ie + 1) & 3] - p0x, d1y = c1y[(ie + 1) & 3] - p0y;
#pragma unroll
            for (int je = 0; je < 4; ++je) {
                float r0x = c2x[je], r0y = c2y[je];
                float d2x = c2x[(je + 1) & 3] - r0x, d2y = c2y[(je + 1) & 3] - r0y;
                float rpx = r0x - p0x, rpy = r0y - p0y;
                float den  = d1x * d2y - d1y * d2x;
                float safe = (fabsf(den) < 1e-12f) ? 1.0f : den;
                float t = (rpx * d2y - rpy * d2x) / safe;
                float u = (rpx * d1y - rpy * d1x) / safe;
                bool valid = (fabsf(den) > 1e-12f) &&
                             (t > 0.0f) && (t < 1.0f) && (u > 0.0f) && (u < 1.0f);
                int idx = 8 + ie * 4 + je;
                vx[idx] = p0x + t * d1x;
                vy[idx] = p0y + t * d1y;
                mk[idx] = valid;
            }
        }

        // convex area (literal port of _convex_area)
        int   nv = 0;
        float mx = 0.0f, my = 0.0f;
#pragma unroll
        for (int k = 0; k < 24; ++k)
            if (mk[k]) { ++nv; mx += vx[k]; my += vy[k]; }
        float inv = 1.0f / (float)(nv > 1 ? nv : 1);
        mx *= inv; my *= inv;

        float sxv[24], syv[24], ang[24];
#pragma unroll
        for (int k = 0; k < 24; ++k) {
            if (mk[k]) {
                sxv[k] = vx[k] - mx; syv[k] = vy[k] - my;
                ang[k] = atan2f(syv[k], sxv[k]);
            } else {
                sxv[k] = 0.0f; syv[k] = 0.0f; ang[k] = 1e9f;
            }
        }
        // stable insertion sort by angle (matches jnp.argsort stability)
        for (int a2 = 1; a2 < 24; ++a2) {
            float ka = ang[a2], kx = sxv[a2], ky = syv[a2];
            int b2 = a2 - 1;
            while (b2 >= 0 && ang[b2] > ka) {
                ang[b2 + 1] = ang[b2]; sxv[b2 + 1] = sxv[b2]; syv[b2 + 1] = syv[b2];
                --b2;
            }
            ang[b2 + 1] = ka; sxv[b2 + 1] = kx; syv[b2 + 1] = ky;
        }
        float crosses = 0.0f;
        for (int k = 0; k < 23; ++k)
            crosses += sxv[k] * syv[k + 1] - sxv[k + 1] * syv[k];
        int last = nv - 1; if (last < 0) last = 0; if (last > 23) last = 23;
        crosses += sxv[last] * syv[0] - sxv[0] * syv[last];
        float inter = 0.5f * fabsf(crosses);

        float a1 = pw * ph, a2r = tw * th;
        float iou = inter / (a1 + a2r - inter + 1e-8f);
        iou = fmaxf(iou, 1e-6f);
        s_iou = (1.0f - iou) * w;

        // ---- DFL ----
        const float* apn = anchor_points + (size_t)(i % L_DIM) * 2;
        float axp = apn[0], ayp = apn[1];
        float dist[4];
        dist[0] = fminf(fmaxf(axp - tx, 0.0f), (float)REG_MAX - 0.01f);
        dist[1] = fminf(fmaxf(ayp - ty, 0.0f), (float)REG_MAX - 0.01f);
        dist[2] = fminf(fmaxf(tw - axp, 0.0f), (float)REG_MAX - 0.01f);
        dist[3] = fminf(fmaxf(th - ayp, 0.0f), (float)REG_MAX - 0.01f);

        float dfl = 0.0f;
#pragma unroll
        for (int ch = 0; ch < 4; ++ch) {
            float t  = dist[ch];
            int   tl = (int)t; if (tl < 0) tl = 0; if (tl > REG_MAX - 1) tl = REG_MAX - 1;
            float wl = (float)(tl + 1) - t;
            float wr = 1.0f - wl;
            const float* lg = pd + ch * (REG_MAX + 1);
            float lgl[REG_MAX + 1];
            float mval = -3.4e38f;
#pragma unroll
            for (int k = 0; k <= REG_MAX; ++k) { lgl[k] = lg[k]; mval = fmaxf(mval, lgl[k]); }
            float se = 0.0f;
#pragma unroll
            for (int k = 0; k <= REG_MAX; ++k) se += __expf(lgl[k] - mval);
            float lse = __logf(se);
            float lpl = lgl[tl]     - mval - lse;
            float lpr = lgl[tl + 1] - mval - lse;
            dfl += -(lpl * wl) - (lpr * wr);
        }
        s_dfl = dfl * 0.25f * w;
    }

    // ---- block tree reduction, then one atomic per block per accumulator ----
    __shared__ float sm[256];
    int t = (int)threadIdx.x;
    sm[t] = s_iou; __syncthreads();
    for (int off = 128; off > 0; off >>= 1) {
        if (t < off) sm[t] += sm[t + off];
        __syncthreads();
    }
    if (t == 0) atomicAdd(&acc[0], sm[0]);
    __syncthreads();
    sm[t] = s_dfl; __syncthreads();
    for (int off = 128; off > 0; off >>= 1) {
        if (t < off) sm[t] += sm[t + off];
        __syncthreads();
    }
    if (t == 0) atomicAdd(&acc[1], sm[0]);
}

// ---------------- Kernel 3: finalize (divide by denom) ---------------------
__global__ void k_finalize(const float* __restrict__ acc,
                           const float* __restrict__ tss,
                           float* __restrict__ out) {
    if (threadIdx.x == 0) {
        float denom = (tss[0] == 0.0f) ? 1.0f : tss[0];
        out[0] = acc[0] / denom;   // loss_iou
        out[1] = acc[1] / denom;   // loss_dfl
    }
}

extern "C" void kernel_launch(void* const* d_in, const int* in_sizes, int n_in,
                              void* d_out, int out_size, void* d_ws, size_t ws_size,
                              hipStream_t stream) {
    const float*         pred_dist     = (const float*)d_in[0];
    const float*         pred_bboxes   = (const float*)d_in[1];
    const float*         pred_angles   = (const float*)d_in[2];
    const float*         anchor_points = (const float*)d_in[3];
    const float*         target_bboxes = (const float*)d_in[4];
    const float*         target_angles = (const float*)d_in[5];
    const float*         target_scores = (const float*)d_in[6];
    const float*         tss           = (const float*)d_in[7];
    const unsigned char* fg_mask       = (const unsigned char*)d_in[8];

    int n = in_sizes[1] / 4;                 // B*L anchors (pred_bboxes is (B,L,4))

    float* acc = (float*)d_ws;               // [0..1] loss accumulators
    float* bw  = acc + 16;                   // bbox_weight, n floats (aligned)

    k_zero<<<1, 32, 0, stream>>>(acc);

    int ntiles = n / 16;                     // n divisible by 16 (B*L = 172032)
    int wblocks = (ntiles + 7) / 8;          // 8 waves (tiles) per 256-thread block
    k_rowsum_wmma<<<wblocks, 256, 0, stream>>>(target_scores, bw, ntiles);

    int mblocks = (n + 255) / 256;
    k_main<<<mblocks, 256, 0, stream>>>(pred_dist, pred_bboxes, pred_angles,
                                        anchor_points, target_bboxes, target_angles,
                                        fg_mask, bw, acc, n);

    k_finalize<<<1, 32, 0, stream>>>(acc, tss, (float*)d_out);
}